// loss_28570122453837
// MI455X (gfx1250) — compile-verified
//
#include <hip/hip_runtime.h>
#include <math.h>

// Problem constants (match reference)
#define NB   8
#define NPTS 2048
#define BDIM 256          // 8 waves of 32
#define NGRP 16           // 16 blocks per batch (128 output points each)
#define NT   (NPTS / 16)  // 128 m-tiles per reduction

// Math constants
#define EPSF     0.2f
#define INV_EPS  5.0f
#define LOG2E_F  1.4426950408889634f
#define LN2_F    0.6931471805599453f
#define K2C      (LOG2E_F * INV_EPS)        // converts nat/EPS -> log2 domain
#define S2C      (2.0f * K2C)               // scale on xy term
#define LOG_AB   (-7.6246189861593985f)     // log(1/2048)

typedef float v2f __attribute__((ext_vector_type(2)));
typedef float v8f __attribute__((ext_vector_type(8)));

// Raw hardware transcendentals: v_exp_f32 flushes deeply-negative inputs to 0,
// which is exactly right inside a logsumexp (no -126 guard sequence needed).
#define EXP2F(x) __builtin_amdgcn_exp2f(x)
#define LOG2F(x) __builtin_amdgcn_logf(x)

// One 16x16 tile of xy = P_tile(16x4,K=3 used) * Q_tile^T(4x16) via fp32 WMMA.
// A 32-bit 16x4 layout: lanes 0-15 hold K=0 (v0), K=1 (v1); lanes 16-31 hold K=2,K=3.
// B 4x16 mirrors. D: lane l<16 -> rows M=v, col N=l; lane l>=16 -> rows M=v+8.
__device__ __forceinline__ v8f xy_tile_wmma(float a0, float a1, float b0, float b1) {
  v2f A; A[0] = a0; A[1] = a1;
  v2f Bv; Bv[0] = b0; Bv[1] = b1;
  v8f Cz = {};
  return __builtin_amdgcn_wmma_f32_16x16x4_f32(false, A, false, Bv, (short)0, Cz,
                                               false, false);
}

// A-operand staging: half-indexed SoA pair table.
//   sh_A[(0*NPTS + i)*2 .. +1] = (x, y)   (lanes 0-15: K=0,1)
//   sh_A[(1*NPTS + i)*2 .. +1] = (z, 0)   (lanes 16-31: K=2,3)
// Each lane's per-tile fetch is ONE unconditional aligned ds_load_b64.
__device__ __forceinline__ void stage_points(float* sh_A, const float* Pb, int i,
                                             float& x, float& y, float& z) {
  x = Pb[i * 3 + 0]; y = Pb[i * 3 + 1]; z = Pb[i * 3 + 2];
  sh_A[(0 * NPTS + i) * 2 + 0] = x;
  sh_A[(0 * NPTS + i) * 2 + 1] = y;
  sh_A[(1 * NPTS + i) * 2 + 0] = z;
  sh_A[(1 * NPTS + i) * 2 + 1] = 0.0f;
}

__device__ __forceinline__ void load_tile(const float* sh_A, const float* sh_t,
                                          int abase, int mt, int half,
                                          float2& av, float4& ta, float4& tb) {
  av = *reinterpret_cast<const float2*>(&sh_A[abase + mt * 32]);
  const float4* t4 = reinterpret_cast<const float4*>(&sh_t[mt * 16 + half * 8]);
  ta = t4[0];
  tb = t4[1];
}

// ---------------------------------------------------------------------------
// Sinkhorn dual half-update:
//   dout[q] = -EPS*log_ab + xx_q - EPS*ln2 * lse2_m( t2[m] + S2C*xy )
// P: points reduced over (with dual din), Q: output-side points.
// ---------------------------------------------------------------------------
__global__ __launch_bounds__(BDIM)
void sinkhorn_dual_kernel(const float* __restrict__ P, const float* __restrict__ Q,
                          const float* __restrict__ din, float* __restrict__ dout) {
  __shared__ __align__(16) float sh_A[NPTS * 4];
  __shared__ __align__(16) float sh_t2[NPTS];

  const int b   = blockIdx.x >> 4;
  const int grp = blockIdx.x & (NGRP - 1);
  const int tid = threadIdx.x;

  const float* Pb = P + (size_t)b * NPTS * 3;
  for (int i = tid; i < NPTS; i += BDIM) {
    float x, y, z;
    stage_points(sh_A, Pb, i, x, y, z);
    sh_t2[i] = (din[b * NPTS + i] - (x * x + y * y + z * z)) * K2C;
  }
  __syncthreads();

  const int wave = tid >> 5, lane = tid & 31;
  const int half = lane >> 4, li = lane & 15;
  const int q = grp * 128 + wave * 16 + li;
  const int abase = (half * NPTS + li) * 2;   // per-lane base into sh_A (floats)

  const float* Qp = Q + ((size_t)b * NPTS + q) * 3;
  const float qx = Qp[0], qy = Qp[1], qz = Qp[2];
  const float xxq = qx * qx + qy * qy + qz * qz;
  const float bb0 = half ? qz : qx;
  const float bb1 = half ? 0.0f : qy;

  float m2 = -__builtin_inff();
  float S  = 0.0f;

  // one-stage software pipeline on LDS operand loads (branch-free wrap index)
  float2 av; float4 ta, tb;
  load_tile(sh_A, sh_t2, abase, 0, half, av, ta, tb);

  for (int mt = 0; mt < NT; ++mt) {
    v8f d = xy_tile_wmma(av.x, av.y, bb0, bb1);
    const float tv[8] = {ta.x, ta.y, ta.z, ta.w, tb.x, tb.y, tb.z, tb.w};

    load_tile(sh_A, sh_t2, abase, (mt + 1) & (NT - 1), half, av, ta, tb);

    // tile-local args (independent FMAs -> v_pk_fma_f32 pairs)
    float a[8];
#pragma unroll
    for (int v = 0; v < 8; ++v) a[v] = fmaf(S2C, d[v], tv[v]);

    // tree max (depth 3, all independent)
    float x01 = fmaxf(a[0], a[1]), x23 = fmaxf(a[2], a[3]);
    float x45 = fmaxf(a[4], a[5]), x67 = fmaxf(a[6], a[7]);
    float tmax = fmaxf(fmaxf(x01, x23), fmaxf(x45, x67));

    // 8 independent exps against the tile max, tree sum
    float e0 = EXP2F(a[0] - tmax), e1 = EXP2F(a[1] - tmax);
    float e2 = EXP2F(a[2] - tmax), e3 = EXP2F(a[3] - tmax);
    float e4 = EXP2F(a[4] - tmax), e5 = EXP2F(a[5] - tmax);
    float e6 = EXP2F(a[6] - tmax), e7 = EXP2F(a[7] - tmax);
    float Et = ((e0 + e1) + (e2 + e3)) + ((e4 + e5) + (e6 + e7));

    // single combine with the running accumulator
    float nm = fmaxf(m2, tmax);
    S  = fmaf(S, EXP2F(m2 - nm), Et * EXP2F(tmax - nm));
    m2 = nm;
  }

  // merge the two half-wave partials (rows 0-7 vs 8-15 of every tile)
  const float m2o = __shfl_xor(m2, 16, 32);
  const float So  = __shfl_xor(S, 16, 32);
  const float nm  = fmaxf(m2, m2o);
  S = S * EXP2F(m2 - nm) + So * EXP2F(m2o - nm);
  const float lse2 = nm + LOG2F(S);
  const float fv = xxq - EPSF * LOG_AB - EPSF * LN2_F * lse2;
  if (half == 0) dout[b * NPTS + q] = fv;
}

// ---------------------------------------------------------------------------
// Chamfer one direction: per output point q, min over P of squared dist.
// Accumulates w * min / NPTS into *out.
// ---------------------------------------------------------------------------
__global__ __launch_bounds__(BDIM)
void chamfer_kernel(const float* __restrict__ P, const float* __restrict__ Q,
                    float w, float* __restrict__ out) {
  __shared__ __align__(16) float sh_A[NPTS * 4];
  __shared__ __align__(16) float sh_yy[NPTS];

  const int b   = blockIdx.x >> 4;
  const int grp = blockIdx.x & (NGRP - 1);
  const int tid = threadIdx.x;

  const float* Pb = P + (size_t)b * NPTS * 3;
  for (int i = tid; i < NPTS; i += BDIM) {
    float x, y, z;
    stage_points(sh_A, Pb, i, x, y, z);
    sh_yy[i] = x * x + y * y + z * z;
  }
  __syncthreads();

  const int wave = tid >> 5, lane = tid & 31;
  const int half = lane >> 4, li = lane & 15;
  const int q = grp * 128 + wave * 16 + li;
  const int abase = (half * NPTS + li) * 2;

  const float* Qp = Q + ((size_t)b * NPTS + q) * 3;
  const float qx = Qp[0], qy = Qp[1], qz = Qp[2];
  const float xxq = qx * qx + qy * qy + qz * qz;
  const float bb0 = half ? qz : qx;
  const float bb1 = half ? 0.0f : qy;

  float minc = __builtin_inff();

  float2 av; float4 ta, tb;
  load_tile(sh_A, sh_yy, abase, 0, half, av, ta, tb);

  for (int mt = 0; mt < NT; ++mt) {
    v8f d = xy_tile_wmma(av.x, av.y, bb0, bb1);
    const float yv[8] = {ta.x, ta.y, ta.z, ta.w, tb.x, tb.y, tb.z, tb.w};

    load_tile(sh_A, sh_yy, abase, (mt + 1) & (NT - 1), half, av, ta, tb);

    float c[8];
#pragma unroll
    for (int v = 0; v < 8; ++v) c[v] = fmaf(-2.0f, d[v], xxq + yv[v]);

    float x01 = fminf(c[0], c[1]), x23 = fminf(c[2], c[3]);
    float x45 = fminf(c[4], c[5]), x67 = fminf(c[6], c[7]);
    float tmin = fminf(fminf(x01, x23), fminf(x45, x67));
    minc = fminf(minc, tmin);
  }

  const float mo = __shfl_xor(minc, 16, 32);
  minc = fminf(minc, mo);
  float contrib = minc * (w * (1.0f / (float)NPTS));
  contrib += __shfl_xor(contrib, 1, 32);
  contrib += __shfl_xor(contrib, 2, 32);
  contrib += __shfl_xor(contrib, 4, 32);
  contrib += __shfl_xor(contrib, 8, 32);
  if (lane == 0) atomicAdd(out, contrib);
}

// ---------------------------------------------------------------------------
// Final EMD pass: sum over all (n,m): exp((f+g-C)/EPS) * C / NPTS
// ---------------------------------------------------------------------------
__global__ __launch_bounds__(BDIM)
void emd_final_kernel(const float* __restrict__ P, const float* __restrict__ Q,
                      const float* __restrict__ gdual, const float* __restrict__ fdual,
                      float* __restrict__ out) {
  __shared__ __align__(16) float sh_A[NPTS * 4];
  __shared__ __align__(16) float sh_yy[NPTS];
  __shared__ __align__(16) float sh_u2[NPTS];

  const int b   = blockIdx.x >> 4;
  const int grp = blockIdx.x & (NGRP - 1);
  const int tid = threadIdx.x;

  const float* Pb = P + (size_t)b * NPTS * 3;
  for (int i = tid; i < NPTS; i += BDIM) {
    float x, y, z;
    stage_points(sh_A, Pb, i, x, y, z);
    float yy = x * x + y * y + z * z;
    sh_yy[i] = yy;
    sh_u2[i] = (gdual[b * NPTS + i] - yy) * K2C;
  }
  __syncthreads();

  const int wave = tid >> 5, lane = tid & 31;
  const int half = lane >> 4, li = lane & 15;
  const int q = grp * 128 + wave * 16 + li;
  const int abase = (half * NPTS + li) * 2;

  const float* Qp = Q + ((size_t)b * NPTS + q) * 3;
  const float qx = Qp[0], qy = Qp[1], qz = Qp[2];
  const float xxq = qx * qx + qy * qy + qz * qz;
  const float bb0 = half ? qz : qx;
  const float bb1 = half ? 0.0f : qy;
  const float fq  = fdual[b * NPTS + q];
  const float c1  = (fq - xxq) * K2C;

  float acc = 0.0f;
  for (int mt = 0; mt < NT; ++mt) {
    const float2 av = *reinterpret_cast<const float2*>(&sh_A[abase + mt * 32]);
    v8f d = xy_tile_wmma(av.x, av.y, bb0, bb1);

    const float4* y4 = reinterpret_cast<const float4*>(&sh_yy[mt * 16 + half * 8]);
    const float4* u4 = reinterpret_cast<const float4*>(&sh_u2[mt * 16 + half * 8]);
    const float4 ya = y4[0], yb = y4[1];
    const float4 ua = u4[0], ub = u4[1];
    const float yv[8] = {ya.x, ya.y, ya.z, ya.w, yb.x, yb.y, yb.z, yb.w};
    const float uv[8] = {ua.x, ua.y, ua.z, ua.w, ub.x, ub.y, ub.z, ub.w};
#pragma unroll
    for (int v = 0; v < 8; ++v) {
      float cost = fmaf(-2.0f, d[v], xxq + yv[v]);
      float e2   = fmaf(S2C, d[v], c1 + uv[v]);
      acc = fmaf(EXP2F(e2), cost, acc);
    }
  }

  acc += __shfl_xor(acc, 1, 32);
  acc += __shfl_xor(acc, 2, 32);
  acc += __shfl_xor(acc, 4, 32);
  acc += __shfl_xor(acc, 8, 32);
  acc += __shfl_xor(acc, 16, 32);
  if (lane == 0) atomicAdd(out, acc * (1.0f / (float)NPTS));
}

// ---------------------------------------------------------------------------
extern "C" void kernel_launch(void* const* d_in, const int* in_sizes, int n_in,
                              void* d_out, int out_size, void* d_ws, size_t ws_size,
                              hipStream_t stream) {
  const float* target = (const float*)d_in[0];  // [B,N,3]
  const float* pre    = (const float*)d_in[1];  // [B,N,3]
  float* out = (float*)d_out;

  float* f = (float*)d_ws;            // dual over pre rows  [B,N]
  float* g = f + NB * NPTS;           // dual over target     [B,M]

  hipMemsetAsync(d_ws, 0, (size_t)2 * NB * NPTS * sizeof(float), stream);
  hipMemsetAsync(d_out, 0, sizeof(float), stream);

  dim3 grid(NB * NGRP), block(BDIM);

  // Chamfer: dist1 (pre->target, w=1), dist2 (target->pre, w=0.5)
  chamfer_kernel<<<grid, block, 0, stream>>>(target, pre, 1.0f, out);
  chamfer_kernel<<<grid, block, 0, stream>>>(pre, target, 0.5f, out);

  // 20 Sinkhorn iterations: f from g, then g from f (global sync = kernel boundary)
  for (int it = 0; it < 20; ++it) {
    sinkhorn_dual_kernel<<<grid, block, 0, stream>>>(target, pre, g, f);
    sinkhorn_dual_kernel<<<grid, block, 0, stream>>>(pre, target, f, g);
  }

  // Final transport-cost accumulation
  emd_final_kernel<<<grid, block, 0, stream>>>(target, pre, g, f, out);
}